// GCN_42434276884907
// MI455X (gfx1250) — compile-verified
//
#include <hip/hip_runtime.h>

// GCN forward for MI455X (gfx1250).
//   - Dense per-node GEMM uses fp32 WMMA: v_wmma_f32_16x16x4_f32 (wave32).
//   - Edge message passing uses native fp32 global atomics (L2-resident:
//     h/agg buffers are 25.6 MB each, well inside the 192 MB L2).
//   - All scratch carved from d_ws; everything launched on `stream`.

#define NNODES   100000
#define NEDGES   1600000
#define FDIM     64            // F_IN == HID == 64
#define NCLASSES 45
#define NGRAPHS  256

typedef __attribute__((ext_vector_type(2))) float v2f;
typedef __attribute__((ext_vector_type(8))) float v8f;

__device__ __forceinline__ void atomic_add_f32(float* p, float v) {
  unsafeAtomicAdd(p, v);     // lowers to global_atomic_add_f32 on gfx1250
}

// ---------------------------------------------------------------- utilities
__global__ void zero_kernel(float* __restrict__ p, long n) {
  long i = (long)blockIdx.x * blockDim.x + threadIdx.x;
  if (i < n) p[i] = 0.0f;
}

__global__ void deg_kernel(const int* __restrict__ dst, float* __restrict__ deg) {
  int e = blockIdx.x * blockDim.x + threadIdx.x;
  if (e < NEDGES) atomic_add_f32(&deg[dst[e]], 1.0f);
}

__global__ void dinv_kernel(float* __restrict__ d) {
  int i = blockIdx.x * blockDim.x + threadIdx.x;
  if (i < NNODES) d[i] = rsqrtf(d[i] + 1.0f);   // deg^{-1/2}, self-loop included
}

// -------------------------------------------------------- WMMA node GEMM
// H[n, 0:64] = X[n, 0:64] @ W[64, 64]  (row-major), fp32 exact.
// One wave -> 16x16 output tile; 4 waves/block cover all 64 columns.
// K loop: 16 steps of v_wmma_f32_16x16x4_f32.
//
// Operand layout (ISA 7.12.2):
//  A (16x4 f32, 2 VGPRs): lanes 0-15 hold M=lane, K={0,1}; lanes 16-31 K={2,3}
//  B (4x16 f32, 2 VGPRs): lanes 0-15 hold N=lane, K={0,1}; lanes 16-31 K={2,3}
//  C/D (16x16 f32, 8 VGPRs): vgpr r = row (r + 8*(lane>>4)), col = lane&15
__global__ void __launch_bounds__(128)
gemm64_wmma(const float* __restrict__ X, const float* __restrict__ W,
            float* __restrict__ H) {
  const int lane = threadIdx.x & 31;
  const int wave = threadIdx.x >> 5;          // 0..3 -> 16-column tile
  const int m    = lane & 15;
  const int half = lane >> 4;                 // 0 -> K pair {0,1}; 1 -> {2,3}
  const int node = blockIdx.x * 16 + m;       // NNODES % 16 == 0
  const int col0 = wave * 16;

  v8f acc = {};
#pragma unroll
  for (int kk = 0; kk < 16; ++kk) {
    const int k = kk * 4 + half * 2;
    v2f a, b;
    a.x = X[node * FDIM + k];
    a.y = X[node * FDIM + k + 1];
    b.x = W[k       * FDIM + col0 + m];
    b.y = W[(k + 1) * FDIM + col0 + m];
    acc = __builtin_amdgcn_wmma_f32_16x16x4_f32(
        /*neg_a=*/false, a, /*neg_b=*/false, b,
        /*c_mod=*/(short)0, acc, /*reuse_a=*/false, /*reuse_b=*/false);
  }

  const int row0 = blockIdx.x * 16 + half * 8;
#pragma unroll
  for (int r = 0; r < 8; ++r)
    H[(row0 + r) * FDIM + col0 + m] = acc[r];
}

// --------------------------------------------------- edge gather + scatter
// One thread per (edge, 4-channel chunk): 16 chunks x 1.6M edges.
// agg[dst] += h[src] * dinv[src]*dinv[dst]
__global__ void edge_scatter(const float4* __restrict__ H4,
                             const int* __restrict__ src,
                             const int* __restrict__ dst,
                             const float* __restrict__ dinv,
                             float* __restrict__ agg, long nwork) {
  long i = (long)blockIdx.x * blockDim.x + threadIdx.x;
  if (i >= nwork) return;
  const int e  = (int)(i >> 4);
  const int c4 = (int)(i & 15);
  const int s = src[e];
  const int d = dst[e];
  const float coef = dinv[s] * dinv[d];
  const float4 v = H4[(size_t)s * 16 + c4];
  float* out = agg + (size_t)d * FDIM + c4 * 4;
  atomic_add_f32(out + 0, v.x * coef);
  atomic_add_f32(out + 1, v.y * coef);
  atomic_add_f32(out + 2, v.z * coef);
  atomic_add_f32(out + 3, v.w * coef);
}

// --------------------------- self-loop contribution + bias (+ReLU), in place
__global__ void finalize_kernel(const float* __restrict__ H,
                                float* __restrict__ agg,
                                const float* __restrict__ dinv,
                                const float* __restrict__ bias, int relu) {
  long i = (long)blockIdx.x * blockDim.x + threadIdx.x;
  if (i >= (long)NNODES * FDIM) return;
  const int node = (int)(i >> 6);
  const int c    = (int)(i & 63);
  const float di = dinv[node];
  float v = agg[i] + H[i] * di * di + bias[c];
  agg[i] = relu ? fmaxf(v, 0.0f) : v;
}

// ------------------------------------------------------------ mean pooling
__global__ void pool_feat_kernel(const float4* __restrict__ H4,
                                 const int* __restrict__ batch,
                                 float* __restrict__ psum) {
  long i = (long)blockIdx.x * blockDim.x + threadIdx.x;  // node*16 + c4
  if (i >= (long)NNODES * 16) return;
  const int node = (int)(i >> 4);
  const int c4   = (int)(i & 15);
  const int g = batch[node];
  const float4 v = H4[i];
  float* out = psum + (size_t)g * FDIM + c4 * 4;
  atomic_add_f32(out + 0, v.x);
  atomic_add_f32(out + 1, v.y);
  atomic_add_f32(out + 2, v.z);
  atomic_add_f32(out + 3, v.w);
}

__global__ void pool_cnt_kernel(const int* __restrict__ batch,
                                float* __restrict__ pcnt) {
  int i = blockIdx.x * blockDim.x + threadIdx.x;
  if (i < NNODES) atomic_add_f32(&pcnt[batch[i]], 1.0f);
}

// ------------------------------------------------------- final classifier
// out[g, c] = (psum[g,:]/max(cnt,1)) @ Wl[:, c] + bl[c]   (256 x 45, tiny)
__global__ void linear_kernel(const float* __restrict__ psum,
                              const float* __restrict__ pcnt,
                              const float* __restrict__ Wl,
                              const float* __restrict__ bl,
                              float* __restrict__ out) {
  const int g = blockIdx.x;
  const int c = threadIdx.x;
  if (c >= NCLASSES) return;
  const float inv = 1.0f / fmaxf(pcnt[g], 1.0f);
  float acc = bl[c];
#pragma unroll 8
  for (int k = 0; k < FDIM; ++k)
    acc = fmaf(psum[g * FDIM + k] * inv, Wl[k * NCLASSES + c], acc);
  out[g * NCLASSES + c] = acc;
}

// -------------------------------------------------------------------- host
static void run_layer(const float* Xin, const float* W, const float* b, int relu,
                      float* Hbuf, float* Agg, const float* dinv,
                      const int* src, const int* dst, hipStream_t stream) {
  const long na = (long)NNODES * FDIM;
  gemm64_wmma<<<NNODES / 16, 128, 0, stream>>>(Xin, W, Hbuf);
  zero_kernel<<<(int)((na + 255) / 256), 256, 0, stream>>>(Agg, na);
  const long nwork = (long)NEDGES * 16;
  edge_scatter<<<(int)((nwork + 255) / 256), 256, 0, stream>>>(
      (const float4*)Hbuf, src, dst, dinv, Agg, nwork);
  finalize_kernel<<<(int)((na + 255) / 256), 256, 0, stream>>>(Hbuf, Agg, dinv, b, relu);
}

extern "C" void kernel_launch(void* const* d_in, const int* in_sizes, int n_in,
                              void* d_out, int out_size, void* d_ws, size_t ws_size,
                              hipStream_t stream) {
  const float* x   = (const float*)d_in[0];
  const int*   ei  = (const int*)d_in[1];
  const int* batch = (const int*)d_in[2];
  const float* W1  = (const float*)d_in[3];
  const float* b1  = (const float*)d_in[4];
  const float* W2  = (const float*)d_in[5];
  const float* b2  = (const float*)d_in[6];
  const float* W3  = (const float*)d_in[7];
  const float* b3  = (const float*)d_in[8];
  const float* Wl  = (const float*)d_in[9];
  const float* bl  = (const float*)d_in[10];
  const int* src = ei;
  const int* dst = ei + NEDGES;

  // Workspace layout (floats): B0,B1,B2 node buffers | dinv | psum | pcnt
  float* B0   = (float*)d_ws;
  float* B1   = B0   + (size_t)NNODES * FDIM;
  float* B2   = B1   + (size_t)NNODES * FDIM;
  float* dinv = B2   + (size_t)NNODES * FDIM;
  float* psum = dinv + NNODES;
  float* pcnt = psum + (size_t)NGRAPHS * FDIM;

  // Degrees -> deg^{-1/2} (computed in place in `dinv`)
  zero_kernel<<<(NNODES + 255) / 256, 256, 0, stream>>>(dinv, NNODES);
  deg_kernel<<<(NEDGES + 255) / 256, 256, 0, stream>>>(dst, dinv);
  dinv_kernel<<<(NNODES + 255) / 256, 256, 0, stream>>>(dinv);

  // Three GCN layers (ping-pong buffers; never mutate d_in)
  run_layer(x,  W1, b1, /*relu=*/1, B0, B1, dinv, src, dst, stream);  // out in B1
  run_layer(B1, W2, b2, /*relu=*/1, B0, B2, dinv, src, dst, stream);  // out in B2
  run_layer(B2, W3, b3, /*relu=*/0, B0, B1, dinv, src, dst, stream);  // out in B1

  // Global mean pool + classifier
  const long np = (long)NGRAPHS * FDIM + NGRAPHS;
  zero_kernel<<<(int)((np + 255) / 256), 256, 0, stream>>>(psum, np);  // psum+pcnt contiguous
  const long nw = (long)NNODES * 16;
  pool_feat_kernel<<<(int)((nw + 255) / 256), 256, 0, stream>>>((const float4*)B1, batch, psum);
  pool_cnt_kernel<<<(NNODES + 255) / 256, 256, 0, stream>>>(batch, pcnt);
  linear_kernel<<<NGRAPHS, 64, 0, stream>>>(psum, pcnt, Wl, bl, (float*)d_out);
}